// GraphAntiSymmetricNN_51307679318016
// MI455X (gfx1250) — compile-verified
//
#include <hip/hip_runtime.h>
#include <hip/hip_bf16.h>
#include <math.h>

#define NN      50000
#define EE      600000
#define DD      128
#define OUTC    40
#define OUTPAD  48
#define ITERS   5
#define GAMMA_C 0.1f
#define EPS_C   0.1f

typedef float v2f __attribute__((ext_vector_type(2)));
typedef float v8f __attribute__((ext_vector_type(8)));

// ---------------------------------------------------------------------------
// Build transposed / antisymmetrized weight matrices once.
//   WembT[k*D+n] = Wemb[n*D+k]
//   WlinT[k*D+n] = Wlin[n*D+k]
//   Amat [k*D+n] = Wanti[k*D+n] - Wanti[n*D+k] - gamma*(k==n)   (B for x@A^T)
//   WoutT[k*48+n] = (n<40) ? Wout[n*D+k] : 0
// ---------------------------------------------------------------------------
__global__ __launch_bounds__(256) void prep_weights_kernel(
    const float* __restrict__ Wemb, const float* __restrict__ Wlin,
    const float* __restrict__ Wanti, const float* __restrict__ Wout,
    float* __restrict__ WembT, float* __restrict__ WlinT,
    float* __restrict__ Amat, float* __restrict__ WoutT)
{
    int i = blockIdx.x * 256 + threadIdx.x;
    if (i < DD * DD) {
        int k = i / DD, n = i % DD;
        WembT[i] = Wemb[n * DD + k];
        WlinT[i] = Wlin[n * DD + k];
        float a  = Wanti[i] - Wanti[n * DD + k];
        Amat[i]  = (k == n) ? (a - GAMMA_C) : a;
    }
    if (i < DD * OUTPAD) {
        int k = i / OUTPAD, n = i % OUTPAD;
        WoutT[i] = (n < OUTC) ? Wout[n * DD + k] : 0.0f;
    }
}

__global__ __launch_bounds__(256) void zero_kernel(float4* __restrict__ p, int n4)
{
    int i = blockIdx.x * 256 + threadIdx.x;
    if (i < n4) p[i] = make_float4(0.f, 0.f, 0.f, 0.f);
}

// ---------------------------------------------------------------------------
// Generic WMMA GEMM: Out[m][n] = sum_k X[m][k] * B[k][n] (+ bias[n])
//   X: [M x 128] row-major, B: [128 x ncolsB] row-major.
//   Block = 128 threads (4 waves). Wave w -> output tile
//   (m0 = blockIdx.x*16, n0 = (blockIdx.y*4 + w)*16).
//   B panel [128 x 16] staged in LDS per wave; A frags loaded b64 from global.
// ---------------------------------------------------------------------------
__device__ __forceinline__ v8f wmma_tile_f32(
    const float* __restrict__ X, const float* __restrict__ Blds,
    int m0, int lm, int hi)
{
    v8f c = {0.f, 0.f, 0.f, 0.f, 0.f, 0.f, 0.f, 0.f};
    const float* xbase = X + (size_t)(m0 + lm) * DD;
#pragma unroll
    for (int k = 0; k < DD; k += 4) {
        // A frag (16x4 f32): lanes 0-15 hold K=k,k+1 ; lanes 16-31 hold K=k+2,k+3
        const float2 av = *(const float2*)(xbase + k + hi * 2);
        v2f a; a.x = av.x; a.y = av.y;
        // B frag (4x16 f32): vgpr0 = row k (+2 for hi lanes), vgpr1 = row k+1 (+2)
        v2f b;
        b.x = Blds[(k + 2 * hi) * 16 + lm];
        b.y = Blds[(k + 2 * hi + 1) * 16 + lm];
        c = __builtin_amdgcn_wmma_f32_16x16x4_f32(
                /*neg_a=*/false, a, /*neg_b=*/false, b,
                /*c_mod=*/(short)0, c, /*reuse_a=*/false, /*reuse_b=*/false);
    }
    return c;
}

__global__ __launch_bounds__(128) void wmma_gemm_kernel(
    const float* __restrict__ X, const float* __restrict__ B,
    const float* __restrict__ bias, float* __restrict__ Out,
    int ncolsB, int outLd, int outCols, int hasBias)
{
    __shared__ float Blds[4][DD * 16];
    const int wave = threadIdx.x >> 5;
    const int lane = threadIdx.x & 31;
    const int lm   = lane & 15;
    const int hi   = lane >> 4;
    const int m0   = blockIdx.x * 16;
    const int n0   = (blockIdx.y * 4 + wave) * 16;
    const bool active = (n0 < ncolsB);

    float* bl = &Blds[wave][0];
    if (active) {
        for (int k = lane; k < DD; k += 32) {
            const float4* s = (const float4*)(B + (size_t)k * ncolsB + n0);
            float4* d = (float4*)(bl + k * 16);
            d[0] = s[0]; d[1] = s[1]; d[2] = s[2]; d[3] = s[3];
        }
    }
    __syncthreads();
    if (!active) return;

    v8f c = wmma_tile_f32(X, bl, m0, lm, hi);

    const int ncol = n0 + lm;
    if (ncol < outCols) {
        const float bv = hasBias ? bias[ncol] : 0.0f;
#pragma unroll
        for (int v = 0; v < 8; ++v) {
            const int m = m0 + v + hi * 8;  // C layout: vgpr v, hi half -> M=v+8
            Out[(size_t)m * outLd + ncol] = c[v] + bv;
        }
    }
}

// ---------------------------------------------------------------------------
// Fused update: Acc[m][n] <- X[m][n] + eps*tanh( (X@Amat)[m][n] + Acc[m][n] + b[n] )
// Safe in place: each Acc element is read exactly once, by its owning tile.
// ---------------------------------------------------------------------------
__global__ __launch_bounds__(128) void wmma_update_kernel(
    const float* __restrict__ X, const float* __restrict__ Amat,
    const float* __restrict__ bconv, float* __restrict__ Acc)
{
    __shared__ float Blds[4][DD * 16];
    const int wave = threadIdx.x >> 5;
    const int lane = threadIdx.x & 31;
    const int lm   = lane & 15;
    const int hi   = lane >> 4;
    const int m0   = blockIdx.x * 16;
    const int n0   = (blockIdx.y * 4 + wave) * 16;

    float* bl = &Blds[wave][0];
    for (int k = lane; k < DD; k += 32) {
        const float4* s = (const float4*)(Amat + (size_t)k * DD + n0);
        float4* d = (float4*)(bl + k * 16);
        d[0] = s[0]; d[1] = s[1]; d[2] = s[2]; d[3] = s[3];
    }
    __syncthreads();

    v8f c = wmma_tile_f32(X, bl, m0, lm, hi);

    const int ncol = n0 + lm;
    const float bv = bconv[ncol];
#pragma unroll
    for (int v = 0; v < 8; ++v) {
        const int m = m0 + v + hi * 8;
        const size_t idx = (size_t)m * DD + ncol;
        const float conv = c[v] + Acc[idx] + bv;
        Acc[idx] = X[idx] + EPS_C * tanhf(conv);
    }
}

// ---------------------------------------------------------------------------
// Edge scatter: one wave (32 lanes) per edge; each lane moves a float4.
//   Acc[dst[e]][:] += Neigh[src[e]][:] * w[e]     (global_atomic_add_f32)
// ---------------------------------------------------------------------------
__global__ __launch_bounds__(256) void scatter_kernel(
    const float* __restrict__ Neigh, const int* __restrict__ src,
    const int* __restrict__ dst, const float* __restrict__ w,
    float* __restrict__ Acc, int nEdges)
{
    const int t    = blockIdx.x * 256 + threadIdx.x;
    const int e    = t >> 5;
    const int lane = t & 31;
    if (e >= nEdges) return;
    const int   s  = src[e];
    const int   d  = dst[e];
    const float wt = w[e];
    const float4 v = ((const float4*)(Neigh + (size_t)s * DD))[lane];
    float* ap = Acc + (size_t)d * DD + lane * 4;
    atomicAdd(ap + 0, v.x * wt);
    atomicAdd(ap + 1, v.y * wt);
    atomicAdd(ap + 2, v.z * wt);
    atomicAdd(ap + 3, v.w * wt);
}

// ---------------------------------------------------------------------------
// Host-side orchestration.
// ---------------------------------------------------------------------------
extern "C" void kernel_launch(void* const* d_in, const int* in_sizes, int n_in,
                              void* d_out, int out_size, void* d_ws, size_t ws_size,
                              hipStream_t stream)
{
    const float* x_in   = (const float*)d_in[0];
    const int*   eidx   = (const int*)  d_in[1];   // [2, E] flat: row0=src, row1=dst
    const float* ew     = (const float*)d_in[2];
    const float* Wemb   = (const float*)d_in[3];
    const float* b_emb  = (const float*)d_in[4];
    const float* Wlin   = (const float*)d_in[5];
    const float* Wanti  = (const float*)d_in[6];
    const float* b_conv = (const float*)d_in[7];
    const float* Wout   = (const float*)d_in[8];
    const float* b_out  = (const float*)d_in[9];
    float*       out    = (float*)d_out;

    const int* src = eidx;
    const int* dst = eidx + EE;

    // Workspace layout (floats)
    float* ws    = (float*)d_ws;
    float* buf0  = ws;                              // x ping
    float* buf1  = buf0 + (size_t)NN * DD;          // neigh
    float* buf2  = buf1 + (size_t)NN * DD;          // agg / x pong
    float* WembT = buf2 + (size_t)NN * DD;
    float* WlinT = WembT + DD * DD;
    float* Amat  = WlinT + DD * DD;
    float* WoutT = Amat  + DD * DD;                 // [128 x 48], zero padded

    const int mTiles = NN / 16;                     // 50000/16 = 3125 exact
    dim3 blk128(128), blk256(256);
    dim3 gemmGridD(mTiles, DD / 64);                // 8 n-tiles / 4 waves
    dim3 gemmGridOut(mTiles, 1);                    // 48 cols -> 3 active waves

    // 1. Weight prep
    prep_weights_kernel<<<(DD * DD + 255) / 256, blk256, 0, stream>>>(
        Wemb, Wlin, Wanti, Wout, WembT, WlinT, Amat, WoutT);

    // 2. Embedding: buf0 = x_in @ Wemb^T + b_emb
    wmma_gemm_kernel<<<gemmGridD, blk128, 0, stream>>>(
        x_in, WembT, b_emb, buf0, DD, DD, DD, 1);

    float* cur = buf0;
    float* acc = buf2;
    const int n4  = (NN * DD) / 4;
    const int sgB = (EE * 32 + 255) / 256;

    for (int it = 0; it < ITERS; ++it) {
        // neigh = cur @ Wlin^T
        wmma_gemm_kernel<<<gemmGridD, blk128, 0, stream>>>(
            cur, WlinT, (const float*)nullptr, buf1, DD, DD, DD, 0);
        // acc = 0
        zero_kernel<<<(n4 + 255) / 256, blk256, 0, stream>>>((float4*)acc, n4);
        // acc[dst] += neigh[src] * w
        scatter_kernel<<<sgB, blk256, 0, stream>>>(buf1, src, dst, ew, acc, EE);
        // acc = cur + eps*tanh(cur@A + acc + b_conv)   (in place)
        wmma_update_kernel<<<gemmGridD, blk128, 0, stream>>>(cur, Amat, b_conv, acc);
        // swap ping/pong
        float* t = cur; cur = acc; acc = t;
    }

    // 3. Readout: out = cur @ Wout^T + b_out   (48-padded B, masked store to 40)
    wmma_gemm_kernel<<<gemmGridOut, blk128, 0, stream>>>(
        cur, WoutT, b_out, out, OUTPAD, OUTC, OUTC, 1);
}